// GCN_50577534878112
// MI455X (gfx1250) — compile-verified
//
#include <hip/hip_runtime.h>

typedef float v2f __attribute__((ext_vector_type(2)));
typedef float v8f __attribute__((ext_vector_type(8)));

#define IN_C 128

// ------------------------------------------------------------------
// degree / normalization:  deg = 1 (self loop) + count(col) ; dinv = rsqrt(deg)
// ------------------------------------------------------------------
__global__ void k_init_deg(float* __restrict__ deg, int n) {
    int i = blockIdx.x * blockDim.x + threadIdx.x;
    if (i < n) deg[i] = 1.0f;   // self-loop contribution
}

__global__ void k_count_deg(const int* __restrict__ col, float* __restrict__ deg, int e) {
    int i = blockIdx.x * blockDim.x + threadIdx.x;
    if (i < e) atomicAdd(&deg[col[i]], 1.0f);
}

__global__ void k_rsqrt_inplace(float* __restrict__ d, int n) {
    int i = blockIdx.x * blockDim.x + threadIdx.x;
    if (i < n) d[i] = rsqrtf(d[i]);   // deg >= 1 always
}

// ------------------------------------------------------------------
// zero-pad W3[128x40] -> W3p[128x48] so the GEMM needs no load guards
// ------------------------------------------------------------------
__global__ void k_pad_w(const float* __restrict__ w, float* __restrict__ wp,
                        int cin, int cout, int coutp) {
    int idx = blockIdx.x * blockDim.x + threadIdx.x;
    if (idx >= cin * coutp) return;
    int k = idx / coutp, c = idx - k * coutp;
    wp[idx] = (c < cout) ? w[k * cout + c] : 0.0f;
}

// ------------------------------------------------------------------
// GEMM: C[n x COUTS] = (RELU? max(A,0) : A)[n x 128] @ Wp[128 x COUTP]
//  - COUTP multiple of 16 (padded); COUTS = real store width/stride
//  - one wave per 16-col tile; MT row-tiles per block reuse B fragments
//  - f32 WMMA 16x16x4; A/B frag: lane(half,v) holds K = k0 + 2*half + v
// ------------------------------------------------------------------
template <int COUTP, int COUTS, int MT, bool RELU>
__global__ __launch_bounds__(32 * (COUTP / 16))
void k_gemm(const float* __restrict__ A, const float* __restrict__ Wp,
            float* __restrict__ C, int nrows) {
    constexpr int NW  = COUTP / 16;         // waves = column tiles
    constexpr int NT  = 32 * NW;            // threads per block
    constexpr int LDA = IN_C + 4;           // padded LDS stride (132 dwords)
    constexpr int KS  = IN_C / 4;           // 32 wmma k-steps
    __shared__ float lds[MT * 16 * LDA];

    const int tid  = threadIdx.x;
    const int wave = tid >> 5;
    const int lane = tid & 31;
    const int half = lane >> 4;             // 0: lanes 0-15, 1: lanes 16-31
    const int lid  = lane & 15;

    const long rowBase = (long)blockIdx.x * (MT * 16);
    const int  ncol    = wave * 16 + lid;   // < COUTP by construction

    // ---- B fragments for this wave's 16 columns, all 32 k-steps (reused MT x)
    v2f b[KS];
    {
        const float* wcol = Wp + ncol;
        #pragma unroll
        for (int ks = 0; ks < KS; ++ks) {
            const int k = 4 * ks + 2 * half;
            b[ks].x = wcol[(k + 0) * COUTP];
            b[ks].y = wcol[(k + 1) * COUTP];
        }
    }

    // ---- cooperative load of MT*16 rows of A into padded LDS (optional ReLU)
    for (int i = tid; i < MT * 16 * (IN_C / 4); i += NT) {
        const int r  = i >> 5;              // IN_C/4 == 32
        const int c4 = i & 31;
        long gr = rowBase + r; if (gr >= nrows) gr = nrows - 1;
        float4 v = *(const float4*)(A + gr * IN_C + c4 * 4);
        if (RELU) {
            v.x = fmaxf(v.x, 0.f); v.y = fmaxf(v.y, 0.f);
            v.z = fmaxf(v.z, 0.f); v.w = fmaxf(v.w, 0.f);
        }
        *(float4*)&lds[r * LDA + c4 * 4] = v;
    }
    __syncthreads();

    // ---- MT accumulation passes reusing the B fragments
    #pragma unroll
    for (int mt = 0; mt < MT; ++mt) {
        v8f acc = {};
        const float* arow = &lds[(mt * 16 + lid) * LDA];
        #pragma unroll
        for (int ks = 0; ks < KS; ++ks) {
            const int k = 4 * ks + 2 * half;
            v2f a;
            a.x = arow[k];
            a.y = arow[k + 1];
            // 8 args: (neg_a, A, neg_b, B, c_mod, C, reuse_a, reuse_b)
            acc = __builtin_amdgcn_wmma_f32_16x16x4_f32(false, a, false, b[ks],
                                                        (short)0, acc, false, false);
        }
        #pragma unroll
        for (int j = 0; j < 8; ++j) {       // C/D: VGPR j -> M = j + 8*half
            const long m = rowBase + mt * 16 + j + 8 * half;
            if (COUTS == COUTP) {
                if (m < nrows) C[m * (long)COUTS + ncol] = acc[j];
            } else {
                if (m < nrows && ncol < COUTS) C[m * (long)COUTS + ncol] = acc[j];
            }
        }
    }
}

// ------------------------------------------------------------------
// out[i,:] = dinv[i]^2 * t[i,:]   (self-loop term; also initializes out)
// ------------------------------------------------------------------
template <int COUT>
__global__ void k_selfinit(const float* __restrict__ t, const float* __restrict__ dinv,
                           float* __restrict__ out, int n) {
    constexpr int C4 = COUT / 4;
    const long idx4  = (long)blockIdx.x * blockDim.x + threadIdx.x;
    const long total = (long)n * C4;
    if (idx4 >= total) return;
    const long i = idx4 / C4;
    float w = dinv[i]; w *= w;
    float4 v = ((const float4*)t)[idx4];
    v.x *= w; v.y *= w; v.z *= w; v.w *= w;
    ((float4*)out)[idx4] = v;
}

// ------------------------------------------------------------------
// one wave per edge: out[col,:] += dinv[row]*dinv[col] * t[row,:]
// ------------------------------------------------------------------
template <int COUT>
__global__ void k_edge_scatter(const float* __restrict__ t,
                               const int* __restrict__ row,
                               const int* __restrict__ col,
                               const float* __restrict__ dinv,
                               float* __restrict__ out, int e) {
    const int  lane = threadIdx.x & 31;
    const long ew   = ((long)blockIdx.x * blockDim.x + threadIdx.x) >> 5;
    if (ew >= e) return;
    const int   r = row[ew];          // wave-uniform -> scalar loads
    const int   c = col[ew];
    const float w = dinv[r] * dinv[c];
    if (COUT == 128) {
        const float4 v = *(const float4*)(t + (long)r * 128 + lane * 4);
        float* dst = out + (long)c * 128 + lane * 4;
        atomicAdd(dst + 0, w * v.x);
        atomicAdd(dst + 1, w * v.y);
        atomicAdd(dst + 2, w * v.z);
        atomicAdd(dst + 3, w * v.w);
    } else {
        for (int ch = lane; ch < COUT; ch += 32)
            atomicAdd(out + (long)c * COUT + ch, w * t[(long)r * COUT + ch]);
    }
}

// ------------------------------------------------------------------
extern "C" void kernel_launch(void* const* d_in, const int* in_sizes, int n_in,
                              void* d_out, int out_size, void* d_ws, size_t ws_size,
                              hipStream_t stream) {
    const float* x  = (const float*)d_in[0];
    const int*   ei = (const int*)d_in[1];
    const float* W1 = (const float*)d_in[2];
    const float* W2 = (const float*)d_in[3];
    const float* W3 = (const float*)d_in[4];
    float* out = (float*)d_out;

    const int N = in_sizes[0] / IN_C;       // 100000
    const int E = in_sizes[1] / 2;          // 3200000
    const int* row = ei;                    // edge_index[0]
    const int* col = ei + E;                // edge_index[1]

    char*  ws   = (char*)d_ws;
    float* dinv = (float*)ws;                                        // N floats
    size_t off  = (((size_t)N * 4) + 255) & ~(size_t)255;
    float* bufA = (float*)(ws + off);                                // N*128
    float* bufB = bufA + (size_t)N * IN_C;                           // N*128
    float* W3p  = bufB + (size_t)N * IN_C;                           // 128*48

    constexpr int MT = 4;                   // row tiles per block (64 rows)
    const int gemmBlocks = (N + MT * 16 - 1) / (MT * 16);

    // ---- normalization ----
    k_init_deg     <<<(N + 255) / 256, 256, 0, stream>>>(dinv, N);
    k_count_deg    <<<(E + 255) / 256, 256, 0, stream>>>(col, dinv, E);
    k_rsqrt_inplace<<<(N + 255) / 256, 256, 0, stream>>>(dinv, N);
    k_pad_w        <<<(128 * 48 + 255) / 256, 256, 0, stream>>>(W3, W3p, 128, 40, 48);

    const int edgeBlocks = (int)(((long)E * 32 + 255) / 256);

    // ---- layer 1: t1 = x @ W1 ; agg1 = Ahat t1 ----
    k_gemm<128, 128, MT, false><<<gemmBlocks, 256, 0, stream>>>(x, W1, bufA, N);
    {
        long t4 = (long)N * 32;
        k_selfinit<128><<<(int)((t4 + 255) / 256), 256, 0, stream>>>(bufA, dinv, bufB, N);
    }
    k_edge_scatter<128><<<edgeBlocks, 256, 0, stream>>>(bufA, row, col, dinv, bufB, E);

    // ---- layer 2: t2 = relu(agg1) @ W2 ; agg2 = Ahat t2 ----
    k_gemm<128, 128, MT, true><<<gemmBlocks, 256, 0, stream>>>(bufB, W2, bufA, N);
    {
        long t4 = (long)N * 32;
        k_selfinit<128><<<(int)((t4 + 255) / 256), 256, 0, stream>>>(bufA, dinv, bufB, N);
    }
    k_edge_scatter<128><<<edgeBlocks, 256, 0, stream>>>(bufA, row, col, dinv, bufB, E);

    // ---- layer 3: t3 = relu(agg2) @ W3 (48-padded, store stride 40) ----
    k_gemm<48, 40, MT, true><<<gemmBlocks, 96, 0, stream>>>(bufB, W3p, bufA, N);
    {
        long t4 = (long)N * 10;
        k_selfinit<40><<<(int)((t4 + 255) / 256), 256, 0, stream>>>(bufA, dinv, out, N);
    }
    k_edge_scatter<40><<<edgeBlocks, 256, 0, stream>>>(bufA, row, col, dinv, out, E);
}